// UCBAttention_20933670601027
// MI455X (gfx1250) — compile-verified
//
#include <hip/hip_runtime.h>
#include <cmath>

// ---------------- problem constants ----------------
constexpr int Bb   = 4;
constexpr int Nn   = 1025;
constexpr int Cc   = 768;
constexpr int Hh   = 12;
constexpr int Dh   = 64;
constexpr int C3   = 3 * Cc;          // 2304
constexpr int Mrows = Bb * Nn;        // 4100
constexpr int NT   = (Nn + 15) / 16;  // 65 tiles of 16 along N
constexpr int KSEL = 256;             // max(1, int(1024*0.25))
constexpr float SCALE = 0.125f;       // Dh^-0.5
constexpr float BETA  = 1.0f;

typedef __attribute__((ext_vector_type(2))) float v2f;
typedef __attribute__((ext_vector_type(8))) float v8f;

// LDS byte address of a generic pointer to __shared__ (for async-to-LDS ops)
__device__ __forceinline__ unsigned lds_addr(const void* p)
{
    return (unsigned)(unsigned long long)(__attribute__((address_space(3))) const void*)p;
}

// Issue one 16-byte async global->LDS transfer (ASYNCcnt-tracked, no VGPR data)
__device__ __forceinline__ void async_b128(const float* gptr, const float* lptr)
{
    asm volatile("global_load_async_to_lds_b128 %0, %1, off"
                 :: "v"(lds_addr(lptr)),
                    "v"((unsigned long long)(const void*)gptr)
                 : "memory");
}

__device__ __forceinline__ void wait_async0()
{
    asm volatile("s_wait_asynccnt 0" ::: "memory");
}

// ---------------- fp32 WMMA GEMM: C[M,N] = A[M,K] @ B[K,N] (+bias) ----------
// block = 128 threads (4 waves). Block tile: 64 rows x 64 cols of C.
// wave w owns rows [64*bx + 16w, +16), all 64 cols -> 4 v8f accumulators.
// grid.x = ceil(M/64), grid.y = N/64   (K % 64 == 0, N % 64 == 0)
__global__ __launch_bounds__(128)
void gemm_wmma_f32(const float* __restrict__ A, const float* __restrict__ Bm,
                   const float* __restrict__ bias, float* __restrict__ C,
                   int M, int K, int N)
{
    // stride 68 floats = 272B: 16B-aligned rows, bank = (4m + k) % 64 (conflict-free)
    __shared__ __attribute__((aligned(16))) float Als[64 * 68];
    __shared__ __attribute__((aligned(16))) float Bls[64 * 64];

    const int tid  = threadIdx.x;
    const int wave = tid >> 5;
    const int lane = tid & 31;
    const int half = lane >> 4;
    const int ml   = lane & 15;
    const int m0   = blockIdx.x * 64;
    const int n0   = blockIdx.y * 64;

    v8f acc[4];
#pragma unroll
    for (int t = 0; t < 4; ++t) acc[t] = (v8f){0.f,0.f,0.f,0.f,0.f,0.f,0.f,0.f};

#pragma unroll 1
    for (int k0 = 0; k0 < K; k0 += 64) {
        // async-stage A 64x64 and B 64x64, one b128 (4 floats) per transfer
#pragma unroll
        for (int i = tid; i < 64 * 16; i += 128) {
            int r = i >> 4, c4 = (i & 15) * 4;
            int gr = min(m0 + r, M - 1);                       // row-clamped
            async_b128(A + (size_t)gr * K + k0 + c4, &Als[r * 68 + c4]);
        }
#pragma unroll
        for (int i = tid; i < 64 * 16; i += 128) {
            int r = i >> 4, c4 = (i & 15) * 4;
            async_b128(Bm + (size_t)(k0 + r) * N + n0 + c4, &Bls[r * 64 + c4]);
        }
        wait_async0();
        __syncthreads();

        const int arow = (wave * 16 + ml) * 68;
#pragma unroll
        for (int kk = 0; kk < 16; ++kk) {
            const int kb = kk * 4 + 2 * half;  // A/B frag: lanes 0-15 K={0,1}, 16-31 K={2,3}
            v2f a;
            a.x = Als[arow + kb];
            a.y = Als[arow + kb + 1];
#pragma unroll
            for (int t = 0; t < 4; ++t) {
                v2f b;
                b.x = Bls[kb * 64 + t * 16 + ml];
                b.y = Bls[(kb + 1) * 64 + t * 16 + ml];
                acc[t] = __builtin_amdgcn_wmma_f32_16x16x4_f32(false, a, false, b,
                                                               (short)0, acc[t], false, false);
            }
        }
        __syncthreads();
    }

#pragma unroll
    for (int t = 0; t < 4; ++t) {
        const int nc = n0 + t * 16 + ml;
        const float bv = bias ? bias[nc] : 0.0f;
#pragma unroll
        for (int r = 0; r < 8; ++r) {
            int mrow = m0 + wave * 16 + r + 8 * half;  // D layout: reg r <-> row r + 8*(lane/16)
            if (mrow < M) C[(size_t)mrow * N + nc] = acc[t][r] + bv;
        }
    }
}

// ---------------- S tile: 16x16 scores = scale * Q_tile @ K_tile^T ----------
__device__ __forceinline__
v8f s_tile(const float* __restrict__ qkv, int b, int h, int i0, int j0, int lane)
{
    const int half = lane >> 4;
    const int ml   = lane & 15;
    const int iq = min(i0 + ml, Nn - 1);
    const int jk = min(j0 + ml, Nn - 1);
    const float* qrow = qkv + (size_t)(b * Nn + iq) * C3 + h * Dh;        // Q
    const float* krow = qkv + (size_t)(b * Nn + jk) * C3 + Cc + h * Dh;   // K (B = K^T)

    v8f s = {0.f, 0.f, 0.f, 0.f, 0.f, 0.f, 0.f, 0.f};
#pragma unroll
    for (int kk = 0; kk < 16; ++kk) {
        const int kb = kk * 4 + 2 * half;
        v2f a, bb;
        a.x  = qrow[kb];  a.y  = qrow[kb + 1];
        bb.x = krow[kb];  bb.y = krow[kb + 1];
        s = __builtin_amdgcn_wmma_f32_16x16x4_f32(false, a, false, bb,
                                                  (short)0, s, false, false);
    }
#pragma unroll
    for (int r = 0; r < 8; ++r) s[r] *= SCALE;
    return s;
}

// ---------------- pass 1: row max/sum (online softmax) + column sums -------
// grid = (NT, H, B), block = 32 (one wave owns 16 query rows)
__global__ __launch_bounds__(32)
void attn_stats(const float* __restrict__ qkv, float* __restrict__ mbuf,
                float* __restrict__ lbuf, float* __restrict__ colsum)
{
    const int lane = threadIdx.x;
    const int half = lane >> 4;
    const int ml   = lane & 15;
    const int i0 = blockIdx.x * 16, h = blockIdx.y, b = blockIdx.z;

    float mrun[8], lrun[8];
#pragma unroll
    for (int r = 0; r < 8; ++r) { mrun[r] = -3.0e38f; lrun[r] = 0.f; }

#pragma unroll 1
    for (int jt = 0; jt < NT; ++jt) {
        v8f s = s_tile(qkv, b, h, i0, jt * 16, lane);
        const bool colok = (jt * 16 + ml) < Nn;
#pragma unroll
        for (int r = 0; r < 8; ++r) {
            float sv = colok ? s[r] : -3.0e38f;
            float tmax = sv;
            for (int o = 1; o < 16; o <<= 1) tmax = fmaxf(tmax, __shfl_xor(tmax, o));
            float mnew = fmaxf(mrun[r], tmax);
            float e  = __expf(sv - mnew);          // invalid cols underflow to 0
            float ts = e;
            for (int o = 1; o < 16; o <<= 1) ts += __shfl_xor(ts, o);
            lrun[r] = lrun[r] * __expf(mrun[r] - mnew) + ts;
            mrun[r] = mnew;
        }
    }

    if (ml == 0) {
#pragma unroll
        for (int r = 0; r < 8; ++r) {
            int row = i0 + r + 8 * half;
            if (row < Nn) {
                mbuf[(size_t)(b * Hh + h) * Nn + row] = mrun[r];
                lbuf[(size_t)(b * Hh + h) * Nn + row] = lrun[r];
            }
        }
    }

    // sweep 2: accumulate column sums of softmax probabilities
#pragma unroll 1
    for (int jt = 0; jt < NT; ++jt) {
        v8f s = s_tile(qkv, b, h, i0, jt * 16, lane);
        const bool colok = (jt * 16 + ml) < Nn;
        float cp = 0.f;
#pragma unroll
        for (int r = 0; r < 8; ++r) {
            int row = i0 + r + 8 * half;
            float p = (colok && row < Nn) ? __expf(s[r] - mrun[r]) / lrun[r] : 0.f;
            cp += p;
        }
        cp += __shfl_xor(cp, 16);                  // combine row halves (same column)
        if (half == 0 && colok)
            atomicAdd(&colsum[b * Nn + jt * 16 + ml], cp);
    }
}

// ---------------- UCB scoring + exact top-k keep mask ----------------------
// grid = B, block = 1024 (one thread per candidate token)
__global__ __launch_bounds__(1024)
void ucb_select(const float* __restrict__ colsum, const float* __restrict__ ucb_count,
                const int* __restrict__ counter_p, float* __restrict__ keep,
                float* __restrict__ cnt)
{
    __shared__ float u[1024];
    const int b  = blockIdx.x;
    const int j  = threadIdx.x;      // candidate index 0..1023
    const int tj = j + 1;            // token index (skip CLS)

    const float logc = __logf((float)(*counter_p) + 1.0f);
    float ex = 0.f;
#pragma unroll
    for (int h = 0; h < Hh; ++h)
        ex += sqrtf(logc / (ucb_count[h * Nn + tj] + 1e-6f));
    ex *= BETA / (float)Hh;

    const float ps = colsum[b * Nn + tj] / ((float)Hh * (float)Nn);
    u[j] = ps + ex;
    __syncthreads();

    const float mine = u[j];
    int rank = 0;
    for (int q = 0; q < 1024; ++q) {
        float v = u[q];
        rank += (v > mine) || (v == mine && q < j);   // stable tie-break like top_k
    }
    const int sel = rank < KSEL;
    keep[b * Nn + tj] = sel ? 1.f : 0.f;
    if (j == 0) keep[b * Nn] = 1.f;                   // CLS always kept
    if (sel) atomicAdd(&cnt[tj], 1.0f);
}

__global__ void score_delta_k(const float* __restrict__ cnt, float* __restrict__ sd)
{
    int idx = blockIdx.x * blockDim.x + threadIdx.x;
    if (idx < Hh * Nn) sd[idx] = cnt[idx % Nn] * (1.0f / (float)Bb);
}

// ---------------- pass 2: pruned, renormalized attention @ V ---------------
// grid = (NT, H, B), block = 32
__global__ __launch_bounds__(32)
void attn_pass2(const float* __restrict__ qkv, const float* __restrict__ mbuf,
                const float* __restrict__ lbuf, const float* __restrict__ keep,
                float* __restrict__ ctx)
{
    __shared__ float Pl[16 * 17];    // P tile staging (D-layout -> A-layout)
    const int lane = threadIdx.x;
    const int half = lane >> 4;
    const int ml   = lane & 15;
    const int i0 = blockIdx.x * 16, h = blockIdx.y, b = blockIdx.z;

    float mi[8], li[8], ki[8], dsum[8];
#pragma unroll
    for (int r = 0; r < 8; ++r) {
        int row = min(i0 + r + 8 * half, Nn - 1);
        mi[r] = mbuf[(size_t)(b * Hh + h) * Nn + row];
        li[r] = 1.0f / lbuf[(size_t)(b * Hh + h) * Nn + row];
        ki[r] = keep[b * Nn + row];
        dsum[r] = 0.f;
    }
    v8f acc[4];
#pragma unroll
    for (int t = 0; t < 4; ++t) acc[t] = (v8f){0.f,0.f,0.f,0.f,0.f,0.f,0.f,0.f};

#pragma unroll 1
    for (int jt = 0; jt < NT; ++jt) {
        v8f s = s_tile(qkv, b, h, i0, jt * 16, lane);
        const int jg = jt * 16 + ml;
        const bool colok = jg < Nn;
        const float kj = colok ? keep[b * Nn + jg] : 0.f;
#pragma unroll
        for (int r = 0; r < 8; ++r) {
            int row = i0 + r + 8 * half;
            float p = 0.f;
            if (colok && row < Nn) {
                p = __expf(s[r] - mi[r]) * li[r];
                if (!(ki[r] > 0.5f || kj > 0.5f)) p = 0.f;   // mask = keep_i OR keep_j
            }
            dsum[r] += p;
            Pl[(r + 8 * half) * 17 + ml] = p;
        }
        __syncthreads();

        // P(16x16) @ V(16x64): 4 output sub-tiles, K in 4 steps of 4
#pragma unroll
        for (int t = 0; t < 4; ++t) {
#pragma unroll
            for (int kk = 0; kk < 4; ++kk) {
                const int kb = kk * 4 + 2 * half;
                v2f a, bv;
                a.x = Pl[ml * 17 + kb];
                a.y = Pl[ml * 17 + kb + 1];
                const int vr0 = min(jt * 16 + kb,     Nn - 1);
                const int vr1 = min(jt * 16 + kb + 1, Nn - 1);
                bv.x = qkv[(size_t)(b * Nn + vr0) * C3 + 2 * Cc + h * Dh + t * 16 + ml];
                bv.y = qkv[(size_t)(b * Nn + vr1) * C3 + 2 * Cc + h * Dh + t * 16 + ml];
                acc[t] = __builtin_amdgcn_wmma_f32_16x16x4_f32(false, a, false, bv,
                                                               (short)0, acc[t], false, false);
            }
        }
        __syncthreads();
    }

    float rscale[8];
#pragma unroll
    for (int r = 0; r < 8; ++r) {
        float d = dsum[r];
        for (int o = 1; o < 16; o <<= 1) d += __shfl_xor(d, o);
        rscale[r] = 1.0f / (d + 1e-8f);
    }
#pragma unroll
    for (int t = 0; t < 4; ++t)
#pragma unroll
        for (int r = 0; r < 8; ++r) {
            int row = i0 + r + 8 * half;
            if (row < Nn)
                ctx[(size_t)(b * Nn + row) * Cc + h * Dh + t * 16 + ml] = acc[t][r] * rscale[r];
        }
}

// ---------------- host-side orchestration ----------------------------------
extern "C" void kernel_launch(void* const* d_in, const int* in_sizes, int n_in,
                              void* d_out, int out_size, void* d_ws, size_t ws_size,
                              hipStream_t stream)
{
    const float* x         = (const float*)d_in[0];
    const float* ucb_count = (const float*)d_in[1];
    const float* Wqkv      = (const float*)d_in[2];
    const float* Wproj     = (const float*)d_in[3];
    const float* bproj     = (const float*)d_in[4];
    const int*   counter   = (const int*)d_in[5];
    // d_in[6] = ucb_enabled (always 1 for this workload)

    float* out = (float*)d_out;
    float* ws  = (float*)d_ws;

    // workspace layout (floats): ~50.8 MB total
    float* qkv    = ws;                                     // 4100*2304
    float* ctx    = qkv + (size_t)Mrows * C3;               // 4100*768
    float* mbuf   = ctx + (size_t)Mrows * Cc;               // B*H*N
    float* lbuf   = mbuf + (size_t)Bb * Hh * Nn;            // B*H*N
    float* colsum = lbuf + (size_t)Bb * Hh * Nn;            // B*N
    float* keep   = colsum + (size_t)Bb * Nn;               // B*N
    float* cnt    = keep + (size_t)Bb * Nn;                 // N

    // zero accumulators (colsum, keep, cnt are contiguous)
    hipMemsetAsync(colsum, 0, (size_t)(2 * Bb * Nn + Nn) * sizeof(float), stream);

    // 1) qkv = x @ W_qkv                  [4100 x 768] @ [768 x 2304]
    gemm_wmma_f32<<<dim3((Mrows + 63) / 64, C3 / 64), 128, 0, stream>>>(
        x, Wqkv, nullptr, qkv, Mrows, Cc, C3);

    // 2) flash pass 1: softmax stats + patch column scores
    attn_stats<<<dim3(NT, Hh, Bb), 32, 0, stream>>>(qkv, mbuf, lbuf, colsum);

    // 3) UCB top-k selection -> keep mask + selection counts
    ucb_select<<<Bb, 1024, 0, stream>>>(colsum, ucb_count, counter, keep, cnt);

    // 4) score_delta -> tail of d_out
    score_delta_k<<<(Hh * Nn + 255) / 256, 256, 0, stream>>>(
        cnt, out + (size_t)Bb * Nn * Cc);

    // 5) flash pass 2: pruned, renormalized attention @ V -> context
    attn_pass2<<<dim3(NT, Hh, Bb), 32, 0, stream>>>(qkv, mbuf, lbuf, keep, ctx);

    // 6) out = context @ W_proj + b_proj  [4100 x 768] @ [768 x 768]
    gemm_wmma_f32<<<dim3((Mrows + 63) / 64, Cc / 64), 128, 0, stream>>>(
        ctx, Wproj, bproj, out, Mrows, Cc, Cc);
}